// Model_DTSSOCC_62646392980181
// MI455X (gfx1250) — compile-verified
//
#include <hip/hip_runtime.h>

typedef float v2f __attribute__((ext_vector_type(2)));
typedef float v4f __attribute__((ext_vector_type(4)));
typedef float v8f __attribute__((ext_vector_type(8)));

#define EPSQ 1e-8f
#define ALPHA_HALF 5.0f   // alpha * 0.5

constexpr int Fq = 32, Dq = 128, Cq = 16;
constexpr int HS_STR = 132;   // padded Hs tile row stride (32 x 132 floats)
constexpr int FT_STR = 34;    // padded F_tj^T row stride  (128 x 34 floats)
constexpr int WB_STR = 34;    // padded w^T row stride     (16 x 34 floats)
constexpr int SM_HS   = 0;      // union: hs 32x132 = 4224  /  ftjT 128x34 = 4352
constexpr int SM_INVN = 4352;   // 32
constexpr int SM_IMP  = 4384;   // 32
constexpr int SM_WB   = 4416;   // 16*34 = 544
constexpr int SM_TOT  = 4960;   // 19,840 bytes per wave

__device__ __forceinline__ v8f zero8() {
  v8f z = {0.f,0.f,0.f,0.f,0.f,0.f,0.f,0.f};
  return z;
}

// D = A(16x4 f32) * B(4x16 f32) + C(16x16 f32), fp32 WMMA — exact fp32 math
__device__ __forceinline__ v8f wmma4(v2f a, v2f b, v8f c) {
  return __builtin_amdgcn_wmma_f32_16x16x4_f32(false, a, false, b, (short)0, c,
                                               false, false);
}

// LDS is wave-private here; DS ops are in-order per wave (ISA 7.3), so a
// dscnt drain + compiler barrier is sufficient (no workgroup barrier needed).
__device__ __forceinline__ void lds_fence() {
  __builtin_amdgcn_wave_barrier();
  asm volatile("s_wait_dscnt 0x0" ::: "memory");
  __builtin_amdgcn_wave_barrier();
}

// ---------------------------------------------------------------------------
// Pre-pass: pack W^T and normalized cluster centers into B-fragment order.
//   WP[(k/2)*128 + e] = { W[e][k], W[e][k+1] }   (k even)  -> 64 KB
//   CP[(k/2)*16  + c] = { Cn[c][k], Cn[c][k+1] } (k even)  ->  8 KB
// Lane reads float2 at ((2s+half)*N + n): fully coalesced per half-wave.
// ---------------------------------------------------------------------------
__global__ void __launch_bounds__(256) dtss_prep(const float* __restrict__ W,
                                                 const float* __restrict__ Cj,
                                                 v2f* __restrict__ WP,
                                                 v2f* __restrict__ CP) {
  if (blockIdx.x < 32) {
    int p  = blockIdx.x * 256 + threadIdx.x;   // p = kh*128 + e, kh in [0,64)
    int e  = p & 127;
    int k  = (p >> 7) * 2;
    v2f v; v.x = W[e * 128 + k]; v.y = W[e * 128 + k + 1];
    WP[p] = v;
  } else {
    __shared__ float cin[16];
    int t = threadIdx.x;
    if (t < 16) {
      float s = 0.f;
      for (int d = 0; d < 128; ++d) { float x = Cj[t * 128 + d]; s += x * x; }
      cin[t] = 1.0f / fmaxf(__builtin_sqrtf(s), EPSQ);
    }
    __syncthreads();
    for (int p = t; p < 1024; p += 256) {      // p = kh*16 + c, kh in [0,64)
      int c = p & 15;
      int k = (p >> 4) * 2;
      v2f v;
      v.x = Cj[c * 128 + k]     * cin[c];
      v.y = Cj[c * 128 + k + 1] * cin[c];
      CP[p] = v;
    }
  }
}

// ---------------------------------------------------------------------------
// Main fused kernel: one wave32 per (b,s) token.
// ---------------------------------------------------------------------------
__global__ void __launch_bounds__(32) dtss_main(const float* __restrict__ Hs,
                                                const float* __restrict__ importance,
                                                const float* __restrict__ bvec,
                                                const v2f* __restrict__ WP,
                                                const v2f* __restrict__ CP,
                                                float* __restrict__ outF,
                                                float* __restrict__ outI) {
  __shared__ __align__(16) float sm[SM_TOT];
  const int token = blockIdx.x;
  const int lane  = threadIdx.x;
  const int half  = lane >> 4;
  const int lq    = lane & 15;

  // -- Phase 1: stage Hs tile [32 x 128] + importance[32] into LDS (coalesced)
  {
    const float* src = Hs + (size_t)token * (Fq * Dq);
    #pragma unroll 4
    for (int j = 0; j < 32; ++j) {
      v4f v = *(const v4f*)(src + j * Dq + lane * 4);
      *(v4f*)&sm[SM_HS + j * HS_STR + lane * 4] = v;
    }
    sm[SM_IMP + lane] = importance[(size_t)token * Fq + lane];
  }
  lds_fence();

  // -- Phase 2: inverse row norms (lane L owns feature row L)
  {
    float acc = 0.f;
    #pragma unroll 8
    for (int j = 0; j < 32; ++j) {
      v4f v = *(const v4f*)&sm[SM_HS + lane * HS_STR + j * 4];
      acc += v.x * v.x + v.y * v.y + v.z * v.z + v.w * v.w;
    }
    sm[SM_INVN + lane] = 1.0f / fmaxf(__builtin_sqrtf(acc), EPSQ);
  }
  lds_fence();

  // -- Phase 3: cos[32x16] = Hn * Cn^T via f32 WMMA (K=128 -> 32 steps)
  v8f cacc0 = zero8(), cacc1 = zero8();
  {
    const float s0 = sm[SM_INVN + lq];        // A-frag row m = lq (tile 0)
    const float s1 = sm[SM_INVN + 16 + lq];   // tile 1
    for (int s = 0; s < 32; ++s) {
      const int k = 4 * s + 2 * half;         // A layout: lanes>=16 carry K+2
      v2f a0 = *(const v2f*)&sm[SM_HS + lq * HS_STR + k];
      v2f a1 = *(const v2f*)&sm[SM_HS + (16 + lq) * HS_STR + k];
      a0 *= s0;
      a1 *= s1;
      v2f bf = CP[(2 * s + half) * 16 + lq];
      cacc0 = wmma4(a0, bf, cacc0);
      cacc1 = wmma4(a1, bf, cacc1);
    }
  }

  // C/D layout: element (lane, vgpr r) of tile t => f = t*16 + r + 8*half, c = lq
  float lg[2][8];
  #pragma unroll
  for (int r = 0; r < 8; ++r) {
    lg[0][r] = ALPHA_HALF * cacc0[r] - ALPHA_HALF;   // -alpha*0.5*(1-cos)
    lg[1][r] = ALPHA_HALF * cacc1[r] - ALPHA_HALF;
  }

  // -- Phase 4a: w3 = softmax over centers (c spans 16 lanes of a half)
  float w3v[2][8], wv[2][8];
  #pragma unroll
  for (int t = 0; t < 2; ++t) {
    #pragma unroll
    for (int r = 0; r < 8; ++r) {
      float m = lg[t][r];
      m = fmaxf(m, __shfl_xor(m, 1, 32));
      m = fmaxf(m, __shfl_xor(m, 2, 32));
      m = fmaxf(m, __shfl_xor(m, 4, 32));
      m = fmaxf(m, __shfl_xor(m, 8, 32));
      float e = __expf(lg[t][r] - m);
      float ss = e;
      ss += __shfl_xor(ss, 1, 32);
      ss += __shfl_xor(ss, 2, 32);
      ss += __shfl_xor(ss, 4, 32);
      ss += __shfl_xor(ss, 8, 32);
      w3v[t][r] = e / ss;
    }
  }

  // -- Phase 4b: w = softmax over features (f spans VGPRs + the two halves)
  {
    float mf = -3.4e38f;
    #pragma unroll
    for (int t = 0; t < 2; ++t)
      #pragma unroll
      for (int r = 0; r < 8; ++r) mf = fmaxf(mf, lg[t][r]);
    mf = fmaxf(mf, __shfl_xor(mf, 16, 32));
    float sf = 0.f;
    #pragma unroll
    for (int t = 0; t < 2; ++t)
      #pragma unroll
      for (int r = 0; r < 8; ++r) { wv[t][r] = __expf(lg[t][r] - mf); sf += wv[t][r]; }
    sf += __shfl_xor(sf, 16, 32);
    const float rs = 1.0f / sf;
    #pragma unroll
    for (int t = 0; t < 2; ++t)
      #pragma unroll
      for (int r = 0; r < 8; ++r) wv[t][r] *= rs;
  }

  // -- Phase 4c: imp = softmax_c( sum_f w3[f][c] * importance[f] )
  {
    float si = 0.f;
    #pragma unroll
    for (int t = 0; t < 2; ++t)
      #pragma unroll
      for (int r = 0; r < 8; ++r)
        si += w3v[t][r] * sm[SM_IMP + t * 16 + 8 * half + r];
    si += __shfl_xor(si, 16, 32);
    float mi = si;
    mi = fmaxf(mi, __shfl_xor(mi, 1, 32));
    mi = fmaxf(mi, __shfl_xor(mi, 2, 32));
    mi = fmaxf(mi, __shfl_xor(mi, 4, 32));
    mi = fmaxf(mi, __shfl_xor(mi, 8, 32));
    float ei = __expf(si - mi);
    float ssi = ei;
    ssi += __shfl_xor(ssi, 1, 32);
    ssi += __shfl_xor(ssi, 2, 32);
    ssi += __shfl_xor(ssi, 4, 32);
    ssi += __shfl_xor(ssi, 8, 32);
    if (half == 0) outI[(size_t)token * Cq + lq] = ei / ssi;
  }

  // -- Phase 5: stash w^T into LDS as next A matrix: wbuf[c][f], stride 34
  #pragma unroll
  for (int t = 0; t < 2; ++t) {
    #pragma unroll
    for (int r = 0; r < 8; r += 2) {
      v2f v; v.x = wv[t][r]; v.y = wv[t][r + 1];
      *(v2f*)&sm[SM_WB + lq * WB_STR + t * 16 + 8 * half + r] = v;
    }
  }
  lds_fence();

  // -- Phase 6: F_tj[32x128] = Hs * W^T, accumulators live in 128 VGPRs
  v8f fac[2][8];
  #pragma unroll
  for (int mt = 0; mt < 2; ++mt)
    #pragma unroll
    for (int nt = 0; nt < 8; ++nt) fac[mt][nt] = zero8();
  for (int s = 0; s < 32; ++s) {
    const int k = 4 * s + 2 * half;
    v2f a0 = *(const v2f*)&sm[SM_HS + lq * HS_STR + k];
    v2f a1 = *(const v2f*)&sm[SM_HS + (16 + lq) * HS_STR + k];
    const v2f* wrow = WP + (2 * s + half) * 128 + lq;   // L2/WGP$-hot
    #pragma unroll
    for (int nt = 0; nt < 8; ++nt) {
      v2f bf = wrow[nt * 16];
      fac[0][nt] = wmma4(a0, bf, fac[0][nt]);
      fac[1][nt] = wmma4(a1, bf, fac[1][nt]);
    }
  }
  lds_fence();  // hs tile fully consumed; about to overwrite it

  // -- Phase 7: relu(F_tj + b) -> LDS transposed: ftjT[e][f], stride 34
  #pragma unroll
  for (int nt = 0; nt < 8; ++nt) {
    const int e = nt * 16 + lq;
    const float bias = bvec[e];
    #pragma unroll
    for (int mt = 0; mt < 2; ++mt) {
      const int f0 = mt * 16 + 8 * half;
      #pragma unroll
      for (int r = 0; r < 8; r += 2) {
        v2f v;
        v.x = fmaxf(fac[mt][nt][r]     + bias, 0.f);
        v.y = fmaxf(fac[mt][nt][r + 1] + bias, 0.f);
        *(v2f*)&sm[SM_HS + e * FT_STR + f0 + r] = v;
      }
    }
  }
  lds_fence();

  // -- Phase 8: f_t_new[16x128] = w^T * F_tj  (K=32 -> 8 WMMA steps)
  v8f g[8];
  #pragma unroll
  for (int nt = 0; nt < 8; ++nt) g[nt] = zero8();
  for (int s = 0; s < 8; ++s) {
    const int k = 4 * s + 2 * half;
    v2f a = *(const v2f*)&sm[SM_WB + lq * WB_STR + k];
    #pragma unroll
    for (int nt = 0; nt < 8; ++nt) {
      v2f bf = *(const v2f*)&sm[SM_HS + (nt * 16 + lq) * FT_STR + k];
      g[nt] = wmma4(a, bf, g[nt]);
    }
  }

  // -- Phase 9: store f_t_new (half-wave coalesced 64B per instruction)
  float* dst = outF + (size_t)token * (Cq * Dq);
  #pragma unroll
  for (int nt = 0; nt < 8; ++nt) {
    const int e = nt * 16 + lq;
    #pragma unroll
    for (int r = 0; r < 8; ++r) dst[(r + 8 * half) * Dq + e] = g[nt][r];
  }
}

extern "C" void kernel_launch(void* const* d_in, const int* in_sizes, int n_in,
                              void* d_out, int out_size, void* d_ws, size_t ws_size,
                              hipStream_t stream) {
  const float* Hs  = (const float*)d_in[0];   // [8,1024,32,128]
  const float* Cj  = (const float*)d_in[1];   // [16,128]
  const float* W   = (const float*)d_in[2];   // [128,128]
  const float* bv  = (const float*)d_in[3];   // [128]
  const float* imp = (const float*)d_in[4];   // [8,1024,32]

  float* outF = (float*)d_out;                        // [8,1024,16,128]
  float* outI = outF + (size_t)8 * 1024 * 16 * 128;   // [8,1024,16]

  v2f* WP = (v2f*)d_ws;                               // 64 KB
  v2f* CP = (v2f*)((char*)d_ws + 8192 * sizeof(v2f)); //  8 KB

  dtss_prep<<<33, 256, 0, stream>>>(W, Cj, WP, CP);
  dtss_main<<<8192, 32, 0, stream>>>(Hs, imp, bv, WP, CP, outF, outI);
}